// AdaptiveSpectralFeatureRefinementEuclidean_72739566125922
// MI455X (gfx1250) — compile-verified
//
#include <hip/hip_runtime.h>
#include <hip/hip_bf16.h>
#include <stdint.h>

typedef float    v2f __attribute__((ext_vector_type(2)));
typedef float    v8f __attribute__((ext_vector_type(8)));
typedef uint32_t v4u __attribute__((ext_vector_type(4)));
typedef int      v4i __attribute__((ext_vector_type(4)));
typedef int      v8i __attribute__((ext_vector_type(8)));

#define BATCH 8
#define CCH   64
#define HH    128
#define WW    128
#define WTILE 64
#define WPAD  66   // WTILE + 2 halo columns

__global__ __launch_bounds__(64)
void asfr_kernel(const float* __restrict__ fe, const float* __restrict__ fu,
                 float* __restrict__ out) {
  __shared__ float fuS[3 * CCH * WPAD];   // [row r][channel c][padded col]
  __shared__ float fuSq[3 * WPAD];        // per staged column ||fu||^2 over C
  __shared__ float scr[2][256];           // per-wave WMMA D bounce buffer

  const int tid  = threadIdx.x;
  const int lane = tid & 31;
  const int wv   = tid >> 5;
  const int half = lane >> 4;             // which 16-lane half of the wave
  const int pl   = lane & 15;

  int blk = blockIdx.x;
  const int tx = blk & 1;                 // which 64-wide tile of the row
  blk >>= 1;
  const int h0 = blk & (HH - 1);
  const int b  = blk >> 7;
  const int wbase = tx * WTILE;

  const int wl = wv * 32 + lane;          // local pixel 0..63
  const int w  = wbase + wl;              // global pixel column

  const size_t chanStride = (size_t)HH * WW;
  const float* feRow = fe + ((size_t)b * CCH * HH + h0) * WW;   // + c*chanStride + w
  const float* fuImg = fu + (size_t)b * CCH * chanStride;       // + c*chanStride + h*W + w

  const int hlo    = (h0 == 0) ? 0 : h0 - 1;
  const int rstart = (h0 == 0) ? 1 : 0;
  const int nrows  = (h0 == 0 || h0 == HH - 1) ? 2 : 3;

  // ---- zero-fill out-of-image rows (reference zero-pads) ----
  if (h0 == 0)      for (int i = tid; i < CCH * WPAD; i += 64) fuS[i] = 0.f;
  if (h0 == HH - 1) for (int i = tid; i < CCH * WPAD; i += 64) fuS[2 * CCH * WPAD + i] = 0.f;

  // ---- halo columns (padded cols 0 and 65) for valid rows ----
  for (int i = tid; i < nrows * CCH * 2; i += 64) {
    int side = i & 1;
    int rc   = i >> 1;
    int c    = rc % CCH;
    int r    = rstart + rc / CCH;
    int gw   = wbase + (side ? WTILE : -1);
    int gh   = hlo + (r - rstart);
    float v  = 0.f;
    if (gw >= 0 && gw < WW) v = fuImg[(size_t)c * chanStride + (size_t)gh * WW + gw];
    fuS[(r * CCH + c) * WPAD + (side ? (WPAD - 1) : 0)] = v;
  }

  // ---- TDM: async 3D tensor tile (64 cols x 64 ch x nrows) -> LDS, pitch 66 via pad ----
  if (wv == 0) {
    uint64_t gaddr = (uint64_t)(uintptr_t)(fuImg + (size_t)hlo * WW + wbase);
    uint32_t laddr = (uint32_t)(uintptr_t)(&fuS[(rstart * CCH) * WPAD + 1]);
    v4u g0;
    g0[0] = 1u;                                                 // count=1, user mode
    g0[1] = laddr;                                              // lds_addr
    g0[2] = (uint32_t)gaddr;                                    // global_addr lo
    g0[3] = (uint32_t)((gaddr >> 32) & 0x1FFFFFFu) | (2u << 30);// addr hi | type=2
    v8i g1;
    g1[0] = (2 << 16) | (1 << 20) | (5 << 22) | (1 << 25);      // f32, pad_en, 64dw interval, 2dw pad
    g1[1] = WTILE << 16;                                        // tensor_dim0
    g1[2] = CCH << 16;                                          // tensor_dim1
    g1[3] = WTILE << 16;                                        // tile_dim0
    g1[4] = CCH | (nrows << 16);                                // tile_dim1 | tile_dim2
    g1[5] = (int)chanStride;                                    // tensor_dim0_stride (c stride)
    g1[6] = WW << 16;                                           // tensor_dim1_stride (h stride)
    g1[7] = 0;
    v4i g2; g2[0] = 3; g2[1] = 0; g2[2] = 0; g2[3] = 0;         // tensor_dim2
    v4i g3; g3[0] = 0; g3[1] = 0; g3[2] = 0; g3[3] = 0;
    v8i gx; gx[0]=0; gx[1]=0; gx[2]=0; gx[3]=0; gx[4]=0; gx[5]=0; gx[6]=0; gx[7]=0;
    __builtin_amdgcn_tensor_load_to_lds(g0, g1, g2, g3, gx, 0);
    __builtin_amdgcn_s_wait_tensorcnt(0);
  }
  __syncthreads();

  // ---- ||fu||^2 per staged (row, padded col) ----
  for (int i = tid; i < 3 * WPAD; i += 64) {
    int r = i / WPAD, wp = i % WPAD;
    const float* p = &fuS[(r * CCH) * WPAD + wp];
    float s = 0.f;
    for (int c = 0; c < CCH; ++c) { float v = p[c * WPAD]; s += v * v; }
    fuSq[i] = s;
  }
  __syncthreads();

  // ---- ||fe||^2 for this lane's pixel ----
  float fesq = 0.f;
  for (int c = 0; c < CCH; ++c) { float v = feRow[(size_t)c * chanStride + w]; fesq += v * v; }

  // ---- cross terms <fe(p), fu(p+delta)> via f32 WMMA 16x16x4, K-chained over C=64 ----
  float cross[3][3];
  for (int t = 0; t < 2; ++t) {
    const int w0l = wv * 32 + t * 16;      // subtile base, local col
    const int w0g = wbase + w0l;           // subtile base, global col
    v8f acc[6];
    const v8f vzero = {0.f, 0.f, 0.f, 0.f, 0.f, 0.f, 0.f, 0.f};
    for (int i = 0; i < 6; ++i) acc[i] = vzero;
    for (int kk = 0; kk < 16; ++kk) {
      const int c0 = kk * 4 + 2 * half;    // this lane's K channels (A layout: K=2*(lane/16)+vgpr)
      v2f a;
      a.x = feRow[(size_t)(c0 + 0) * chanStride + (w0g + pl)];
      a.y = feRow[(size_t)(c0 + 1) * chanStride + (w0g + pl)];
      for (int dh = 0; dh < 3; ++dh) {
        const float* b0 = &fuS[(dh * CCH + c0) * WPAD];
        for (int cb = 0; cb < 2; ++cb) {
          const int col = w0l + pl + 16 * cb;   // B column j = N + 16*cb, padded col = w0l + j
          v2f bb;
          bb.x = b0[col];
          bb.y = b0[WPAD + col];
          acc[dh * 2 + cb] = __builtin_amdgcn_wmma_f32_16x16x4_f32(
              false, a, false, bb, (short)0, acc[dh * 2 + cb], false, false);
        }
      }
    }
    // D[p][j] band extraction (j = p+1+dw) through per-wave LDS bounce
    float* ms = scr[wv];
    for (int dh = 0; dh < 3; ++dh) {
      #pragma unroll
      for (int r = 0; r < 8; ++r) ms[(r + 8 * half) * 16 + pl] = acc[dh * 2 + 0][r];
      float m_m1 = ms[pl * 16 + pl];                      // dw=-1 -> j=p
      float m_0  = (pl <= 14) ? ms[pl * 16 + pl + 1] : 0.f;
      float m_p1 = (pl <= 13) ? ms[pl * 16 + pl + 2] : 0.f;
      #pragma unroll
      for (int r = 0; r < 8; ++r) ms[(r + 8 * half) * 16 + pl] = acc[dh * 2 + 1][r];
      if (pl == 15) m_0  = ms[15 * 16 + 0];               // j=16 -> tail col 0
      if (pl >= 14) m_p1 = ms[pl * 16 + (pl - 14)];       // j=16,17 -> tail cols 0,1
      if (t == half) { cross[dh][0] = m_m1; cross[dh][1] = m_0; cross[dh][2] = m_p1; }
    }
  }

  // ---- distances (Gram expansion) + softmax over 9 neighbors ----
  float wgt[3][3];
  float dmin = 3.4e38f;
  for (int dh = 0; dh < 3; ++dh)
    for (int di = 0; di < 3; ++di) {
      float d2 = fuSq[dh * WPAD + wl + di] - 2.f * cross[dh][di] + fesq;
      float d  = sqrtf(fmaxf(d2, 0.f));
      wgt[dh][di] = d;
      dmin = fminf(dmin, d);
    }
  float sum = 0.f;
  for (int dh = 0; dh < 3; ++dh)
    for (int di = 0; di < 3; ++di) {
      float e = __expf(dmin - wgt[dh][di]);
      wgt[dh][di] = e;
      sum += e;
    }
  const float inv = 1.f / sum;
  for (int dh = 0; dh < 3; ++dh)
    for (int di = 0; di < 3; ++di) wgt[dh][di] *= inv;

  // ---- weighted 9-tap gather from LDS + residual, store ----
  float* op = out + ((size_t)b * CCH * HH + h0) * WW;
  for (int c = 0; c < CCH; ++c) {
    float accv = feRow[(size_t)c * chanStride + w];
    for (int dh = 0; dh < 3; ++dh) {
      const float* rp = &fuS[(dh * CCH + c) * WPAD + wl];
      accv += wgt[dh][0] * rp[0] + wgt[dh][1] * rp[1] + wgt[dh][2] * rp[2];
    }
    op[(size_t)c * chanStride + w] = accv;
  }
}

extern "C" void kernel_launch(void* const* d_in, const int* in_sizes, int n_in,
                              void* d_out, int out_size, void* d_ws, size_t ws_size,
                              hipStream_t stream) {
  const float* fe = (const float*)d_in[0];   // fe_lv   [8,64,128,128] f32
  const float* fu = (const float*)d_in[1];   // fused   [8,64,128,128] f32
  float* out = (float*)d_out;                // refined [8,64,128,128] f32
  dim3 grid(BATCH * HH * 2);
  dim3 block(64);
  asfr_kernel<<<grid, block, 0, stream>>>(fe, fu, out);
}